// BoxLoss2_6854767805207
// MI455X (gfx1250) — compile-verified
//
#include <hip/hip_runtime.h>
#include <hip/hip_bf16.h>

// CIoU box loss for MI455X (gfx1250, wave32).
// Memory-bound streaming kernel (~168 MB traffic -> ~7us at 23.3 TB/s):
// 128-bit NT loads, NT map stores, speculative GL2 prefetch (unguarded,
// branch-free), WMMA (V_WMMA_F32_16X16X4_F32) ones-trick wave reduction,
// two-stage block-partial reduction (no contended atomics).

typedef __attribute__((ext_vector_type(4))) float f4;
typedef __attribute__((ext_vector_type(2))) float v2f;
typedef __attribute__((ext_vector_type(8))) float v8f;

#define EPSF 1e-9f
#define INV_PI2_4 0.405284734569351085775517852838911f /* 4 / pi^2 */

#define NBLOCKS 2048
#define NTHREADS 256
#define NWAVES (NTHREADS / 32)

// Sum x over all 32 lanes of the wave using one WMMA.
// B = all-ones is layout-invariant; A has x in VGPR0, 0 in VGPR1:
//   lane l<16:  A[M=l][K=0] = x_l          lane l>=16: A[M=l-16][K=2] = x_l
// => D[m][*] = x_m + x_(m+16). Lanes 0-15 hold rows 0..7 in their 8 D VGPRs,
// lanes 16-31 hold rows 8..15; sum 8 VGPRs then shfl_xor(16).
__device__ __forceinline__ float wave_sum_wmma(float x) {
    v2f a; a.x = x;    a.y = 0.0f;
    v2f b; b.x = 1.0f; b.y = 1.0f;
    v8f c = {0.f, 0.f, 0.f, 0.f, 0.f, 0.f, 0.f, 0.f};
    v8f d = __builtin_amdgcn_wmma_f32_16x16x4_f32(
        /*neg_a=*/false, a, /*neg_b=*/false, b,
        /*c_mod=*/(short)0, c, /*reuse_a=*/false, /*reuse_b=*/false);
    float s = d[0] + d[1] + d[2] + d[3] + d[4] + d[5] + d[6] + d[7];
    s += __shfl_xor(s, 16, 32);
    return s;
}

__global__ void __launch_bounds__(NTHREADS)
ciou_kernel(const f4* __restrict__ P, const f4* __restrict__ T,
            const int* __restrict__ M, float* __restrict__ outmap,
            float* __restrict__ partials, int n) {
    const int tid    = blockIdx.x * blockDim.x + threadIdx.x;
    const int stride = gridDim.x * blockDim.x;

    float num = 0.0f;   // sum of (1 - ciou) * mask
    float cnt = 0.0f;   // sum of mask

#pragma unroll 2
    for (int i = tid; i < n; i += stride) {
        // Speculative GL2 prefetch one stride ahead. SYS-scope prefetch is
        // speculative per the ISA (translation failures silently dropped),
        // so no bounds guard -> no per-iteration branch.
        __builtin_prefetch(&P[i + stride], 0, 0);
        __builtin_prefetch(&T[i + stride], 0, 0);

        f4 pb = __builtin_nontemporal_load(&P[i]);  // x1a y1a x2a y2a
        f4 tb = __builtin_nontemporal_load(&T[i]);  // x1b y1b x2b y2b
        float m = (__builtin_nontemporal_load(&M[i]) != 0) ? 1.0f : 0.0f;

        float x1a = pb.x, y1a = pb.y, x2a = pb.z, y2a = pb.w;
        float x1b = tb.x, y1b = tb.y, x2b = tb.z, y2b = tb.w;

        // intersection / union
        float iw    = fminf(x2a, x2b) - fmaxf(x1a, x1b);
        float ih    = fminf(y2a, y2b) - fmaxf(y1a, y1b);
        float inter = fmaxf(iw, 0.0f) * fmaxf(ih, 0.0f);
        float area1 = (x2a - x1a) * (y2a - y1a);
        float area2 = (x2b - x1b) * (y2b - y1b);
        float uni   = area1 + area2 - inter;
        float iou   = inter / (uni + EPSF);

        // center distance over enclosing diagonal
        float dx   = (x1a + x2a) - (x1b + x2b);
        float dy   = (y1a + y2a) - (y1b + y2b);
        float cd   = 0.25f * dx * dx + 0.25f * dy * dy;
        float cw   = fmaxf(x2a, x2b) - fminf(x1a, x1b);
        float ch   = fmaxf(y2a, y2b) - fminf(y1a, y1b);
        float diag = cw * cw + ch * ch + EPSF;
        float diou = iou - cd / diag;

        // aspect-ratio consistency
        float at    = atanf((x2a - x1a) / (y2a - y1a + EPSF)) -
                      atanf((x2b - x1b) / (y2b - y1b + EPSF));
        float v     = INV_PI2_4 * at * at;
        float alpha = v / (v - iou + 1.0f + EPSF);
        float ciou  = diou - alpha * v;

        __builtin_nontemporal_store(ciou * m, &outmap[i]);
        num += (1.0f - ciou) * m;
        cnt += m;
    }

    // Wave32 reduction on the matrix pipe (EXEC reconverged to all-ones here).
    float wnum = wave_sum_wmma(num);
    float wcnt = wave_sum_wmma(cnt);

    __shared__ float snum[NWAVES];
    __shared__ float scnt[NWAVES];
    const int lane = threadIdx.x & 31;
    const int wave = threadIdx.x >> 5;
    if (lane == 0) { snum[wave] = wnum; scnt[wave] = wcnt; }
    __syncthreads();
    if (threadIdx.x == 0) {
        float a = 0.0f, b = 0.0f;
        for (int w = 0; w < NWAVES; ++w) { a += snum[w]; b += scnt[w]; }
        partials[2 * blockIdx.x]     = a;
        partials[2 * blockIdx.x + 1] = b;
    }
}

__global__ void __launch_bounds__(NTHREADS)
finalize_kernel(const float* __restrict__ partials, int nblocks,
                float* __restrict__ loss_out) {
    float num = 0.0f, cnt = 0.0f;
    for (int i = threadIdx.x; i < nblocks; i += blockDim.x) {
        num += partials[2 * i];
        cnt += partials[2 * i + 1];
    }
    for (int off = 1; off < 32; off <<= 1) {
        num += __shfl_xor(num, off, 32);
        cnt += __shfl_xor(cnt, off, 32);
    }
    __shared__ float sn[NWAVES];
    __shared__ float sc[NWAVES];
    const int lane = threadIdx.x & 31;
    const int wave = threadIdx.x >> 5;
    if (lane == 0) { sn[wave] = num; sc[wave] = cnt; }
    __syncthreads();
    if (threadIdx.x == 0) {
        float a = 0.0f, b = 0.0f;
        for (int w = 0; w < NWAVES; ++w) { a += sn[w]; b += sc[w]; }
        loss_out[0] = a / fmaxf(b, 1.0f);
    }
}

extern "C" void kernel_launch(void* const* d_in, const int* in_sizes, int n_in,
                              void* d_out, int out_size, void* d_ws, size_t ws_size,
                              hipStream_t stream) {
    // Inputs: predicts_bbox [B,A,4] f32, targets_bbox [B,A,4] f32,
    //         valid_masks [B,A] int, box_norm [1] f32, cls_norm [1] f32 (unused).
    const f4*  P = (const f4*)d_in[0];
    const f4*  T = (const f4*)d_in[1];
    const int* M = (const int*)d_in[2];
    const int  n = in_sizes[0] / 4;          // number of box pairs (B*A)

    float* out      = (float*)d_out;         // [0] = loss, [1..n] = masked CIoU map
    float* partials = (float*)d_ws;          // 2 floats per block (16 KB)

    ciou_kernel<<<NBLOCKS, NTHREADS, 0, stream>>>(P, T, M, out + 1, partials, n);
    finalize_kernel<<<1, NTHREADS, 0, stream>>>(partials, NBLOCKS, out);
}